// PixelEmbedding2d_64896955843089
// MI455X (gfx1250) — compile-verified
//
#include <hip/hip_runtime.h>
#include <math.h>

typedef __attribute__((ext_vector_type(2))) float v2f;
typedef __attribute__((ext_vector_type(8))) float v8f;

#define HW_TOT   (512u * 512u)   // H*W per (b,c) plane
#define LOG2HW   18
#define WIDTH_   128

// One wave32 handles 16 pixels x 128 channels:
//   A (16x4 f32)  : 16 pixels, K = {x_c0, x_c1, x_c2, 0}
//   B (4x16 f32)  : conv_w^T tile for 16 output channels (8 tiles -> 128 ch)
//   D = A*B       : v_wmma_f32_16x16x4_f32, 8 accumulator tiles
// Epilogue (RoPE + RMS + SiLU) runs directly on the WMMA D layout:
//   lane = channel-in-tile, VGPR row = pixel row; RoPE partner via shfl_xor(1),
//   RMS via shfl_xor butterfly over the 16-lane half.
__global__ __launch_bounds__(256, 2)
void pixemb_rope_rms_silu(const float* __restrict__ x,
                          const float* __restrict__ conv_w,
                          const float* __restrict__ conv_b,
                          const float* __restrict__ norm_w,
                          float* __restrict__ out)
{
    const int lane  = threadIdx.x & 31;
    const int wave  = threadIdx.x >> 5;
    const int lhalf = lane >> 4;        // 0: pixels r .. r ; 1: pixels r+8
    const int l16   = lane & 15;

    const unsigned pixelBase = (blockIdx.x * 8u + (unsigned)wave) * 16u;

    // ---------------- A matrix: 16 pixels x K=4 ----------------
    // ISA layout: lanes 0-15 -> VGPR0=K0, VGPR1=K1 (pixel M=lane)
    //             lanes 16-31 -> VGPR0=K2, VGPR1=K3(=0 pad)
    const unsigned pm = pixelBase + (unsigned)l16;
    const unsigned bb = pm >> LOG2HW;           // batch index
    const unsigned hw = pm & (HW_TOT - 1u);     // flattened h*W+w
    const int cA = lhalf ? 2 : 0;
    const float a0 = x[(size_t)(bb * 3u + (unsigned)cA) * HW_TOT + hw];
    const float a1 = x[(size_t)(bb * 3u + 1u) * HW_TOT + hw];
    v2f A;
    A.x = a0;
    A.y = lhalf ? 0.0f : a1;

    // ------------- B tiles + per-channel constants -------------
    v2f  Bt[8];
    float bias[8], nw[8], invf[8];
#pragma unroll
    for (int t = 0; t < 8; ++t) {
        const int ch = t * 16 + l16;            // output channel 0..127
        const float w0 = conv_w[ch * 3 + cA];
        const float w1 = conv_w[ch * 3 + 1];
        Bt[t].x = w0;
        Bt[t].y = lhalf ? 0.0f : w1;
        bias[t] = conv_b[ch];
        nw[t]   = norm_w[ch];
        const int j = (ch & 63) >> 1;           // RoPE pair index 0..31
        // inv_freq = 10000^(-j/32) = exp(-j * ln(10000)/32)
        invf[t] = __expf(-0.28782313f * (float)j);
    }

    // ---------------- 8x v_wmma_f32_16x16x4_f32 ----------------
    v8f acc[8];
#pragma unroll
    for (int t = 0; t < 8; ++t) {
        v8f cz = {0.f, 0.f, 0.f, 0.f, 0.f, 0.f, 0.f, 0.f};
        acc[t] = __builtin_amdgcn_wmma_f32_16x16x4_f32(
            /*neg_a=*/false, A, /*neg_b=*/false, Bt[t],
            /*c_mod=*/(short)0, cz, /*reuse_a=*/false, /*reuse_b=*/false);
    }

    // -------- pixel positions for each accumulator row ---------
    float posh[8], posw[8];
#pragma unroll
    for (int r = 0; r < 8; ++r) {
        const unsigned p   = pixelBase + (unsigned)r + 8u * (unsigned)lhalf;
        const unsigned hwp = p & (HW_TOT - 1u);
        posh[r] = (float)(hwp >> 9);      // h
        posw[r] = (float)(hwp & 511u);    // w
    }

    // ------------------- bias + 2D RoPE ------------------------
    const float sgn = (lane & 1) ? 1.0f : -1.0f;
    float sumsq[8] = {0.f, 0.f, 0.f, 0.f, 0.f, 0.f, 0.f, 0.f};
#pragma unroll
    for (int t = 0; t < 8; ++t) {
#pragma unroll
        for (int r = 0; r < 8; ++r) {
            const float v       = acc[t][r] + bias[t];
            const float partner = __shfl_xor(v, 1, 32);   // RoPE pair mate
            const float pos     = (t < 4) ? posh[r] : posw[r];
            const float ang     = pos * invf[t];
            const float cs      = __cosf(ang);
            const float sn      = __sinf(ang);
            // even channel: e1*cos - e2*sin ; odd channel: e1*sin + e2*cos
            const float res = v * cs + sgn * sn * partner;
            acc[t][r] = res;
            sumsq[r] += res * res;
        }
    }

    // ------------- RMS norm + SiLU + coalesced store -----------
#pragma unroll
    for (int r = 0; r < 8; ++r) {
        float s = sumsq[r];
        s += __shfl_xor(s, 1, 32);
        s += __shfl_xor(s, 2, 32);
        s += __shfl_xor(s, 4, 32);
        s += __shfl_xor(s, 8, 32);        // sum over the 16-lane half -> 128 ch
        const float inv = __builtin_amdgcn_rsqf(s * (1.0f / 128.0f) + 1e-6f);

        const unsigned p = pixelBase + (unsigned)r + 8u * (unsigned)lhalf;
        float* op = out + (size_t)p * WIDTH_;
#pragma unroll
        for (int t = 0; t < 8; ++t) {
            const float o  = acc[t][r] * inv * nw[t];
            const float so = o * __builtin_amdgcn_rcpf(1.0f + __expf(-o));
            op[t * 16 + l16] = so;        // 16 lanes -> 64B contiguous segment
        }
    }
}

extern "C" void kernel_launch(void* const* d_in, const int* in_sizes, int n_in,
                              void* d_out, int out_size, void* d_ws, size_t ws_size,
                              hipStream_t stream) {
    const float* x      = (const float*)d_in[0];
    const float* conv_w = (const float*)d_in[1];
    const float* conv_b = (const float*)d_in[2];
    const float* norm_w = (const float*)d_in[3];
    float* out = (float*)d_out;

    const int pixels = out_size / WIDTH_;      // 2,097,152
    const int blocks = pixels / 128;           // 128 pixels per 256-thread block
    pixemb_rope_rms_silu<<<dim3(blocks), dim3(256), 0, stream>>>(
        x, conv_w, conv_b, norm_w, out);
}